// ESIM_19799799234746
// MI455X (gfx1250) — compile-verified
//
#include <hip/hip_runtime.h>
#include <cstdint>
#include <cstddef>

#define CP 0.1f
#define CN 0.1f
#define T_FRAMES 16
#define H_DIM 260
#define W_DIM 346
#define HW (H_DIM * W_DIM)          // 89960
#define S_STEPS (T_FRAMES - 1)      // 15
#define NMAX 12
#define N_EVENTS (S_STEPS * HW * NMAX)  // 16,192,800
#define RTILE 4096

typedef __attribute__((ext_vector_type(2))) float v2f;
typedef __attribute__((ext_vector_type(8))) float v8f;

// float -> order-preserving uint key (ascending), and inverse
__device__ __forceinline__ unsigned key_fwd(float t) {
    unsigned u = __float_as_uint(t);
    return (u & 0x80000000u) ? ~u : (u | 0x80000000u);
}
__device__ __forceinline__ float key_inv(unsigned k) {
    unsigned u = (k & 0x80000000u) ? (k ^ 0x80000000u) : ~k;
    return __uint_as_float(u);
}

// ---------------- Stage 1: temporal scan (per pixel, serial over 15 steps) ----
__global__ void esim_scan(const float* __restrict__ video,
                          const float* __restrict__ ts,
                          float* __restrict__ Ac, float* __restrict__ Bc,
                          unsigned char* __restrict__ flags,
                          unsigned* __restrict__ stepCnt) {
    int pix = blockIdx.x * blockDim.x + threadIdx.x;
    bool act = pix < HW;
    int p0 = act ? pix : (HW - 1);
    float ref = video[p0];
    float f0  = ref;
    for (int s = 0; s < S_STEPS; ++s) {
        float f1 = video[(size_t)(s + 1) * HW + p0];
        if (s + 2 < T_FRAMES)
            __builtin_prefetch(&video[(size_t)(s + 2) * HW + p0], 0, 1);
        float t0 = ts[s], t1 = ts[s + 1];
        float dt = t1 - t0;
        bool ev = ((int)(fabsf((f1 - ref) / CP)) > 0);
        unsigned long long bal = __ballot(act && ev);
        if ((threadIdx.x & 31) == 0)
            atomicAdd(&stepCnt[s], (unsigned)__popcll(bal));
        float pf  = (f1 >= f0) ? 1.0f : 0.0f;
        float pol = pf * CP - (1.0f - pf) * CN;
        int numev = (int)((f1 - ref) / pol + 0.001f);
        bool tol  = fabsf(f1 - f0) > 1e-6f;
        float temp = dt / (f1 - f0);
        float Bv = t0 + (ref - f0) * temp;
        float Av = pol * temp;
        if (act) {
            Ac[(size_t)s * HW + pix] = Av;
            Bc[(size_t)s * HW + pix] = Bv;
            int nc = numev < 0 ? 0 : (numev > 13 ? 13 : numev); // clamp; n<=12 comparisons preserved
            flags[(size_t)s * HW + pix] =
                (unsigned char)(nc | (tol ? 16 : 0) | (pf > 0.5f ? 32 : 0));
        }
        ref = ref + (float)numev * pol;
        f0  = f1;
    }
}

// ---------------- Stage 2: expand t = A*n + B via V_WMMA_F32_16X16X4_F32 -----
// Per wave: A-matrix 16x4 rows = [A_pix, B_pix, 0, 0]; B-matrix 4x16 =
// [n_1..n_16 ; 1..1 ; 0 ; 0]  ->  D[16x16] = t_ev for 16 pixels x 16 n-cols.
__global__ void esim_expand(const float* __restrict__ Ac,
                            const float* __restrict__ Bc,
                            const unsigned char* __restrict__ flags,
                            unsigned* __restrict__ keys,
                            unsigned* __restrict__ vals) {
    int s    = blockIdx.y;
    int lane = threadIdx.x & 31;
    int wave = threadIdx.x >> 5;
    int grp  = blockIdx.x * 8 + wave;
    int base = grp * 16;
    if (base >= HW) return;                 // uniform per wave; EXEC stays all-ones

    int row = lane & 15;
    int prc = base + row; prc = prc < HW ? prc : HW - 1;
    float av = Ac[(size_t)s * HW + prc];
    float bv = Bc[(size_t)s * HW + prc];
    bool lowHalf = lane < 16;

    // 32-bit A-matrix 16x4 layout: lanes 0-15 hold K=0,1 (VGPR0,1); lanes 16-31 K=2,3 (zeros)
    v2f a; a.x = lowHalf ? av : 0.0f;       a.y = lowHalf ? bv : 0.0f;
    // 32-bit B-matrix 4x16 layout: lanes 0-15 = rows K=0 (n) / K=1 (1); lanes 16-31 rows K=2,3 (zeros)
    v2f b; b.x = lowHalf ? (float)(row + 1) : 0.0f;  b.y = lowHalf ? 1.0f : 0.0f;
    v8f c = {0.f,0.f,0.f,0.f,0.f,0.f,0.f,0.f};

    v8f d = __builtin_amdgcn_wmma_f32_16x16x4_f32(
        /*neg_a=*/false, a, /*neg_b=*/false, b,
        /*c_mod=*/(short)0, c, /*reuse_a=*/false, /*reuse_b=*/false);

    union { v8f v; float f[8]; } u; u.v = d;
    int col   = lane & 15;                  // n-column index: n = col+1
    int mbase = base + (lowHalf ? 0 : 8);   // C/D layout: VGPR r -> M=r (lanes<16) / M=r+8
    if (col < NMAX) {
        for (int r = 0; r < 8; ++r) {
            int pr = mbase + r;
            if (pr < HW) {
                unsigned char fl = flags[(size_t)s * HW + pr];
                int numev = fl & 15;
                bool tol  = (fl >> 4) & 1;
                bool valid = ((col + 1) <= numev) && tol;
                float t = valid ? u.f[r] : __builtin_huge_valf();
                unsigned e = ((unsigned)(s * HW + pr)) * NMAX + (unsigned)col;
                keys[e] = key_fwd(t);
                vals[e] = e;
            }
        }
    }
}

// ---------------- Stage 3: stable LSD radix sort (4 x 8-bit passes) ----------
__global__ void radix_hist(const unsigned* __restrict__ keys,
                           unsigned* __restrict__ gh,
                           int shift, int numBlocks, int n) {
    __shared__ unsigned h[256];
    h[threadIdx.x] = 0;
    __syncthreads();
    int base = blockIdx.x * RTILE;
    for (int k = 0; k < 16; ++k) {
        int idx = base + k * 256 + threadIdx.x;
        if (idx < n) atomicAdd(&h[(keys[idx] >> shift) & 255u], 1u);
    }
    __syncthreads();
    gh[(size_t)threadIdx.x * numBlocks + blockIdx.x] = h[threadIdx.x];
}

__global__ void radix_offsets(unsigned* __restrict__ gh, int numBlocks) {
    int b = threadIdx.x;
    unsigned running = 0;
    for (int blk = 0; blk < numBlocks; ++blk) {
        size_t i = (size_t)b * numBlocks + blk;
        unsigned t = gh[i]; gh[i] = running; running += t;
    }
    __shared__ unsigned tot[256];
    tot[b] = running;
    __syncthreads();
    for (int off = 1; off < 256; off <<= 1) {   // Hillis-Steele inclusive scan
        unsigned v = (b >= off) ? tot[b - off] : 0;
        __syncthreads();
        tot[b] += v;
        __syncthreads();
    }
    unsigned basev = tot[b] - running;          // exclusive base for bucket b
    for (int blk = 0; blk < numBlocks; ++blk)
        gh[(size_t)b * numBlocks + blk] += basev;
}

// Scatter: stage the key tile into LDS once via CDNA5 async global->LDS copies
// (GLOBAL_LOAD_ASYNC_TO_LDS_B128, ASYNCcnt), then both the counting and the
// scatter phases read keys from LDS — halves global key traffic per pass.
__global__ void radix_scatter(const unsigned* __restrict__ keysIn,
                              const unsigned* __restrict__ valsIn,
                              unsigned* __restrict__ keysOut,
                              unsigned* __restrict__ valsOut,
                              const unsigned* __restrict__ gh,
                              int shift, int numBlocks, int n) {
    __shared__ unsigned lkeys[RTILE];           // 16 KB staged key tile
    __shared__ unsigned wh[8 * 256];            // per-wave bucket counters
    __shared__ unsigned baseOff[256];
    int tid = threadIdx.x, lane = tid & 31, w = tid >> 5;
    int tileBase = blockIdx.x * RTILE;

    // --- async stage: each thread copies 16B x 4 rounds (4096 x 4B total) ---
    {
        unsigned lk = (unsigned)(size_t)(void*)lkeys;   // LDS byte offset (addr[31:0])
        unsigned byteOff = (unsigned)tid * 16u;
        const unsigned* gbase = keysIn + (size_t)tileBase;
        for (int r = 0; r < 4; ++r) {
            unsigned ldsAddr = lk + byteOff + (unsigned)r * 4096u;
            unsigned gOff    = byteOff + (unsigned)r * 4096u;
            asm volatile("global_load_async_to_lds_b128 %0, %1, %2"
                         :: "v"(ldsAddr), "v"(gOff), "s"(gbase)
                         : "memory");
        }
    }
    for (int i = tid; i < 8 * 256; i += 256) wh[i] = 0;  // overlapped with async copy
    asm volatile("s_wait_asynccnt 0x0" ::: "memory");
    __syncthreads();

    // phase 1: per-wave bucket counts (from LDS)
    int chunk = w * 512;                        // contiguous 512 elements per wave
    for (int r = 0; r < 16; ++r) {
        int li = chunk + r * 32 + lane;
        if (tileBase + li < n)
            atomicAdd(&wh[w * 256 + ((lkeys[li] >> shift) & 255u)], 1u);
    }
    __syncthreads();
    // phase 2: exclusive prefix over waves per bucket; load global bases
    {
        unsigned running = 0;
        for (int ww = 0; ww < 8; ++ww) {
            unsigned t = wh[ww * 256 + tid]; wh[ww * 256 + tid] = running; running += t;
        }
        baseOff[tid] = gh[(size_t)tid * numBlocks + blockIdx.x];
    }
    __syncthreads();
    // phase 3: stable scatter — serial rounds per wave, ballot multi-split per round
    for (int r = 0; r < 16; ++r) {
        int li = chunk + r * 32 + lane;
        int idx = tileBase + li;
        bool act = idx < n;
        unsigned key = act ? lkeys[li] : 0u;
        unsigned d = (key >> shift) & 255u;
        unsigned mask = (unsigned)__ballot(act);
        for (int bb = 0; bb < 8; ++bb) {
            unsigned bit = (d >> bb) & 1u;
            unsigned bal = (unsigned)__ballot(bit != 0);
            mask &= bit ? bal : ~bal;
        }
        unsigned rank = __popc(mask & ((1u << lane) - 1u));
        int leader = __ffs((int)mask) - 1;       // -1 if no active lanes match
        unsigned old = 0;
        if (act && lane == leader)
            old = atomicAdd(&wh[w * 256 + d], (unsigned)__popc(mask));
        old = __shfl(old, leader < 0 ? 0 : leader, 32);
        if (act) {
            unsigned pos = baseOff[d] + old + rank;
            keysOut[pos] = key;
            valsOut[pos] = valsIn[idx];
        }
    }
}

// ---------------- Stage 4: decode sorted events + valid mask -----------------
__global__ void esim_gather(const unsigned* __restrict__ keys,
                            const unsigned* __restrict__ vals,
                            const unsigned char* __restrict__ flags,
                            float* __restrict__ out) {
    int i = blockIdx.x * blockDim.x + threadIdx.x;
    if (i >= N_EVENTS) return;
    unsigned k = keys[i], e = vals[i];
    unsigned col = e % NMAX;
    unsigned pe  = e / NMAX;
    unsigned pix = pe % HW;
    unsigned s   = pe / HW;
    unsigned char fl = flags[(size_t)s * HW + pix];
    bool tol  = (fl >> 4) & 1;
    bool pbit = (fl >> 5) & 1;
    unsigned numev = fl & 15u;
    bool valid = ((col + 1u) <= numev) && tol;
    float t = key_inv(k);
    size_t o = (size_t)i * 4;
    out[o + 0] = (float)(pix % W_DIM);
    out[o + 1] = (float)(pix / W_DIM);
    out[o + 2] = t;
    out[o + 3] = pbit ? 1.0f : 0.0f;
    out[(size_t)4 * N_EVENTS + i] = valid ? 1.0f : 0.0f;
}

__global__ void esim_changed(const unsigned* __restrict__ stepCnt,
                             float* __restrict__ out) {
    int s = threadIdx.x;
    if (s < S_STEPS)
        out[(size_t)5 * N_EVENTS + s] =
            ((float)stepCnt[s] >= 0.7f * (float)HW) ? 1.0f : 0.0f;
}

__global__ void init_cnt(unsigned* c) {
    if (threadIdx.x < S_STEPS) c[threadIdx.x] = 0;
}

// ---------------- Launch ------------------------------------------------------
extern "C" void kernel_launch(void* const* d_in, const int* in_sizes, int n_in,
                              void* d_out, int out_size, void* d_ws, size_t ws_size,
                              hipStream_t stream) {
    (void)in_sizes; (void)n_in; (void)out_size; (void)ws_size;
    const float* ts    = (const float*)d_in[0];   // timestamps (16)
    const float* video = (const float*)d_in[1];   // (16, 260, 346)
    float* out = (float*)d_out;

    char* ws = (char*)d_ws;
    size_t o = 0;
    auto alloc = [&](size_t bytes) -> void* {
        void* p = ws + o;
        o += (bytes + 255) & ~(size_t)255;
        return p;
    };
    const int N = N_EVENTS;
    unsigned* keysA = (unsigned*)alloc((size_t)N * 4);
    unsigned* valsA = (unsigned*)alloc((size_t)N * 4);
    unsigned* keysB = (unsigned*)alloc((size_t)N * 4);
    unsigned* valsB = (unsigned*)alloc((size_t)N * 4);
    float* Ac = (float*)alloc((size_t)S_STEPS * HW * 4);
    float* Bc = (float*)alloc((size_t)S_STEPS * HW * 4);
    unsigned char* flags = (unsigned char*)alloc((size_t)S_STEPS * HW);
    const int numBlocks = (N + RTILE - 1) / RTILE;
    unsigned* gh = (unsigned*)alloc((size_t)numBlocks * 256 * 4);
    unsigned* stepCnt = (unsigned*)alloc(64);

    init_cnt<<<1, 32, 0, stream>>>(stepCnt);
    esim_scan<<<(HW + 255) / 256, 256, 0, stream>>>(video, ts, Ac, Bc, flags, stepCnt);

    dim3 eg((HW + 16 * 8 - 1) / (16 * 8), S_STEPS);
    esim_expand<<<eg, 256, 0, stream>>>(Ac, Bc, flags, keysA, valsA);

    unsigned *ki = keysA, *vi = valsA, *ko = keysB, *vo = valsB;
    for (int pass = 0; pass < 4; ++pass) {
        int shift = pass * 8;
        radix_hist<<<numBlocks, 256, 0, stream>>>(ki, gh, shift, numBlocks, N);
        radix_offsets<<<1, 256, 0, stream>>>(gh, numBlocks);
        radix_scatter<<<numBlocks, 256, 0, stream>>>(ki, vi, ko, vo, gh, shift, numBlocks, N);
        unsigned* tk = ki; ki = ko; ko = tk;
        unsigned* tv = vi; vi = vo; vo = tv;
    }
    // after 4 passes result is back in keysA/valsA (ki/vi)
    esim_gather<<<(N + 255) / 256, 256, 0, stream>>>(ki, vi, flags, out);
    esim_changed<<<1, 32, 0, stream>>>(stepCnt, out);
}